// BackWarp_50714973831911
// MI455X (gfx1250) — compile-verified
//
#include <hip/hip_runtime.h>

// dense_image_warp (tfa-style backward warp, bilinear, clamped) for
// image [16,360,640,3] f32, flow [16,360,640,2] f32 -> out [16,360,640,3] f32.
//
// Memory-bound kernel (~120-160 MB total traffic -> ~5-7us at 23.3 TB/s).
// Design: 1 pixel/thread, 3D grid to avoid integer div/mod, float2 NT loads
// for flow, wide (24B) merged loads for the 2x2 texel footprint (regular
// temporal: whole image fits in 192MB L2 so gathers hit), NT stores for the
// write-once output.

typedef float v2f __attribute__((ext_vector_type(2)));

constexpr int Bn = 16;
constexpr int Hn = 360;
constexpr int Wn = 640;
constexpr int Cn = 3;
constexpr int BLOCK_X = 128;   // 4 wave32 waves per workgroup; 640 = 5*128

__global__ __launch_bounds__(BLOCK_X) void dense_warp_kernel(
    const float* __restrict__ img,
    const v2f*   __restrict__ flow,
    float*       __restrict__ out)
{
    const int x = blockIdx.x * BLOCK_X + threadIdx.x;  // 0..639
    const int y = blockIdx.y;                          // 0..359
    const int b = blockIdx.z;                          // 0..15

    const int p = (b * Hn + y) * Wn + x;               // flat pixel index

    // flow[..., 0] = dy, flow[..., 1] = dx  (read-once -> non-temporal)
    const v2f f = __builtin_nontemporal_load(flow + p);
    const float qy = (float)y - f.x;
    const float qx = (float)x - f.y;

    // TFA clamping: floor clipped to [0, size-2], alpha clipped to [0,1]
    float fy = floorf(qy);
    float fx = floorf(qx);
    fy = fminf(fmaxf(fy, 0.0f), (float)(Hn - 2));
    fx = fminf(fmaxf(fx, 0.0f), (float)(Wn - 2));
    const float ay = fminf(fmaxf(qy - fy, 0.0f), 1.0f);
    const float ax = fminf(fmaxf(qx - fx, 0.0f), 1.0f);
    const int iy = (int)fy;
    const int ix = (int)fx;

    // 2x2 texel footprint. tl|tr and bl|br are each 24 contiguous bytes.
    const float* __restrict__ p0 =
        img + (size_t)(((b * Hn + iy) * Wn + ix) * Cn);
    const float* __restrict__ p1 = p0 + (size_t)(Wn * Cn);

    float top2[6];  // tl.rgb, tr.rgb
    float bot2[6];  // bl.rgb, br.rgb
    __builtin_memcpy(top2, p0, 6 * sizeof(float));
    __builtin_memcpy(bot2, p1, 6 * sizeof(float));

    float o[3];
#pragma unroll
    for (int c = 0; c < 3; ++c) {
        const float top = top2[c] + (top2[c + 3] - top2[c]) * ax;
        const float bot = bot2[c] + (bot2[c + 3] - bot2[c]) * ax;
        o[c] = top + (bot - top) * ay;
    }

    // write-once output -> non-temporal stores (don't rinse image from L2)
    float* __restrict__ op = out + (size_t)p * Cn;
    __builtin_nontemporal_store(o[0], op + 0);
    __builtin_nontemporal_store(o[1], op + 1);
    __builtin_nontemporal_store(o[2], op + 2);
}

extern "C" void kernel_launch(void* const* d_in, const int* in_sizes, int n_in,
                              void* d_out, int out_size, void* d_ws, size_t ws_size,
                              hipStream_t stream) {
    (void)in_sizes; (void)n_in; (void)out_size; (void)d_ws; (void)ws_size;

    const float* img  = (const float*)d_in[0];           // [16,360,640,3]
    const v2f*   flow = (const v2f*)d_in[1];             // [16,360,640,2]
    float*       out  = (float*)d_out;                   // [16,360,640,3]

    dim3 block(BLOCK_X, 1, 1);
    dim3 grid(Wn / BLOCK_X, Hn, Bn);                     // (5, 360, 16)
    dense_warp_kernel<<<grid, block, 0, stream>>>(img, flow, out);
}